// MyGNN_50560355008781
// MI455X (gfx1250) — compile-verified
//
#include <hip/hip_runtime.h>
#include <hip/hip_bf16.h>

typedef float v2f __attribute__((ext_vector_type(2)));
typedef float v8f __attribute__((ext_vector_type(8)));

#define NEG_SLOPE 0.2f

// ---------- helpers ----------
__device__ __forceinline__ unsigned fkey(float x) {
    unsigned u = __float_as_uint(x);
    return (u & 0x80000000u) ? ~u : (u | 0x80000000u);   // monotonic order-preserving key
}
__device__ __forceinline__ float funkey(unsigned k) {
    return (k & 0x80000000u) ? __uint_as_float(k & 0x7FFFFFFFu) : __uint_as_float(~k);
}
__device__ __forceinline__ float leaky(float v) { return v > 0.0f ? v : NEG_SLOPE * v; }

// ---------- tiny prep: softmax(alpha), wa_src = W@att_src, wa_dst = W@att_dst ----------
__global__ void gnn_prep(const float* __restrict__ alpha, const float* __restrict__ Wg,
                         const float* __restrict__ att_s, const float* __restrict__ att_d,
                         float* __restrict__ wts, float* __restrict__ wa_s, float* __restrict__ wa_d) {
    int t = threadIdx.x;
    if (t == 0) {
        float m = alpha[0];
        for (int i = 1; i < 4; ++i) m = fmaxf(m, alpha[i]);
        float e[4], s = 0.0f;
        for (int i = 0; i < 4; ++i) { e[i] = expf(alpha[i] - m); s += e[i]; }
        for (int i = 0; i < 4; ++i) wts[i] = e[i] / s;
    }
    if (t < 11) {
        float as = 0.0f, ad = 0.0f;
        for (int d = 0; d < 64; ++d) {
            float w = Wg[t * 64 + d];
            as += w * att_s[d];
            ad += w * att_d[d];
        }
        wa_s[t] = as;
        wa_d[t] = ad;
    }
}

// ---------- zero fill ----------
__global__ void gnn_zero(float* __restrict__ p, long n) {
    long i = (long)blockIdx.x * blockDim.x + threadIdx.x;
    long stride = (long)gridDim.x * blockDim.x;
    for (; i < n; i += stride) p[i] = 0.0f;
}

// ---------- h = x @ W_gat  (K = 11) ----------
__global__ void gnn_xw(const float* __restrict__ x, const float* __restrict__ Wg,
                       float* __restrict__ h, int N) {
    long i = (long)blockIdx.x * blockDim.x + threadIdx.x;
    if (i >= (long)N * 64) return;
    int n = (int)(i >> 6), d = (int)(i & 63);
    float acc = 0.0f;
    #pragma unroll
    for (int k = 0; k < 11; ++k) acc += x[n * 11 + k] * Wg[k * 64 + d];
    h[i] = acc;
}

// ---------- per-node attention scalars: a = x @ (W@att)  ----------
__global__ void gnn_attn(const float* __restrict__ x, const float* __restrict__ wa_s,
                         const float* __restrict__ wa_d, float* __restrict__ a_src,
                         float* __restrict__ a_dst, int N) {
    int n = blockIdx.x * blockDim.x + threadIdx.x;
    if (n >= N) return;
    float as = 0.0f, ad = 0.0f;
    #pragma unroll
    for (int k = 0; k < 11; ++k) {
        float xv = x[n * 11 + k];
        as += xv * wa_s[k];
        ad += xv * wa_d[k];
    }
    a_src[n] = as;
    a_dst[n] = ad;
}

__device__ __forceinline__ void edge_sd(long e, const int* src, const int* dst, long E,
                                        int& s, int& d) {
    if (e < E) { s = src[e]; d = dst[e]; }
    else       { s = (int)(e - E); d = s; }          // self loops
}

// ---------- segment max of logits over dst (atomic on monotonic key) ----------
__global__ void gnn_seg_max(const int* __restrict__ src, const int* __restrict__ dst,
                            const float* __restrict__ a_src, const float* __restrict__ a_dst,
                            unsigned* __restrict__ mkey, long E, int N) {
    long e = (long)blockIdx.x * blockDim.x + threadIdx.x;
    if (e >= E + N) return;
    int s, d; edge_sd(e, src, dst, E, s, d);
    float lg = leaky(a_src[s] + a_dst[d]);
    atomicMax(&mkey[d], fkey(lg));
}

// ---------- segment sum of exp(logit - m) ----------
__global__ void gnn_seg_sum(const int* __restrict__ src, const int* __restrict__ dst,
                            const float* __restrict__ a_src, const float* __restrict__ a_dst,
                            const unsigned* __restrict__ mkey, float* __restrict__ ssum,
                            long E, int N) {
    long e = (long)blockIdx.x * blockDim.x + threadIdx.x;
    if (e >= E + N) return;
    int s, d; edge_sd(e, src, dst, E, s, d);
    float lg = leaky(a_src[s] + a_dst[d]);
    atomicAdd(&ssum[d], expf(lg - funkey(mkey[d])));
}

// ---------- GAT weighted aggregation: one wave per edge, float2 per lane ----------
__global__ void gnn_gat_agg(const int* __restrict__ src, const int* __restrict__ dst,
                            const float* __restrict__ a_src, const float* __restrict__ a_dst,
                            const unsigned* __restrict__ mkey, const float* __restrict__ ssum,
                            const float* __restrict__ h, float* __restrict__ out,
                            long E, int N) {
    long t = (long)blockIdx.x * blockDim.x + threadIdx.x;
    long e = t >> 5;
    int lane = (int)(t & 31);
    if (e >= E + N) return;
    int s, d; edge_sd(e, src, dst, E, s, d);
    float lg = leaky(a_src[s] + a_dst[d]);
    float w = expf(lg - funkey(mkey[d])) / (ssum[d] + 1e-16f);
    int f = lane * 2;
    atomicAdd(&out[(long)d * 64 + f],     w * h[(long)s * 64 + f]);
    atomicAdd(&out[(long)d * 64 + f + 1], w * h[(long)s * 64 + f + 1]);
}

// ---------- add bias; seed output accumulator with weight[0] ----------
__global__ void gnn_bias_seed(float* __restrict__ gat, const float* __restrict__ b,
                              const float* __restrict__ wts, float* __restrict__ outAcc, int N) {
    long i = (long)blockIdx.x * blockDim.x + threadIdx.x;
    if (i >= (long)N * 64) return;
    int d = (int)(i & 63);
    float g = gat[i] + b[d];
    gat[i] = g;
    outAcc[i] = g * wts[0];
}

// ---------- degree (no self loops for SAGE) ----------
__global__ void gnn_deg(const int* __restrict__ dst, float* __restrict__ deg, long E) {
    long e = (long)blockIdx.x * blockDim.x + threadIdx.x;
    if (e >= E) return;
    atomicAdd(&deg[dst[e]], 1.0f);
}

// ---------- SAGE sum aggregation: wave per edge ----------
__global__ void gnn_sage_agg(const int* __restrict__ src, const int* __restrict__ dst,
                             const float* __restrict__ xin, float* __restrict__ agg, long E) {
    long t = (long)blockIdx.x * blockDim.x + threadIdx.x;
    long e = t >> 5;
    int lane = (int)(t & 31);
    if (e >= E) return;
    int s = src[e], d = dst[e];
    int f = lane * 2;
    atomicAdd(&agg[(long)d * 64 + f],     xin[(long)s * 64 + f]);
    atomicAdd(&agg[(long)d * 64 + f + 1], xin[(long)s * 64 + f + 1]);
}

// ---------- divide by degree (in place) ----------
__global__ void gnn_mean(float* __restrict__ agg, const float* __restrict__ deg, int N) {
    long i = (long)blockIdx.x * blockDim.x + threadIdx.x;
    if (i >= (long)N * 64) return;
    agg[i] = agg[i] / fmaxf(deg[i >> 6], 1.0f);
}

// ---------- fused SAGE GEMM via V_WMMA_F32_16X16X4_F32 ----------
// out = mean @ Wl + hin @ Wr + b ; outAcc += out * wts[widx]
// block = 128 threads = 4 waves; each wave owns one 16-wide column tile of a 16-row slab.
__global__ void gnn_sage_gemm(const float* __restrict__ mean, const float* __restrict__ hin,
                              const float* __restrict__ Wl, const float* __restrict__ Wr,
                              const float* __restrict__ bias, const float* __restrict__ wts,
                              int widx, float* __restrict__ outS, float* __restrict__ outAcc,
                              int N) {
    const int lane = threadIdx.x & 31;
    const int wave = threadIdx.x >> 5;           // column tile 0..3
    const int row0 = blockIdx.x * 16;
    const int col0 = wave * 16;
    const int half = lane >> 4;                  // 0: K+0/1 ; 1: K+2/3
    const int l16  = lane & 15;
    int arow = row0 + l16;
    if (arow >= N) arow = N - 1;                 // clamp loads; stores predicated below
    const int kh = half * 2;

    v8f acc = {};
    #pragma unroll
    for (int k = 0; k < 64; k += 4) {
        v2f a, b;
        // chain 1: mean @ Wl
        a.x = mean[(long)arow * 64 + k + kh];
        a.y = mean[(long)arow * 64 + k + kh + 1];
        b.x = Wl[(k + kh) * 64 + col0 + l16];
        b.y = Wl[(k + kh + 1) * 64 + col0 + l16];
        acc = __builtin_amdgcn_wmma_f32_16x16x4_f32(false, a, false, b, (short)0, acc,
                                                    false, false);
        // chain 2: hin @ Wr
        a.x = hin[(long)arow * 64 + k + kh];
        a.y = hin[(long)arow * 64 + k + kh + 1];
        b.x = Wr[(k + kh) * 64 + col0 + l16];
        b.y = Wr[(k + kh + 1) * 64 + col0 + l16];
        acc = __builtin_amdgcn_wmma_f32_16x16x4_f32(false, a, false, b, (short)0, acc,
                                                    false, false);
    }
    const float w  = wts[widx];
    const float bv = bias[col0 + l16];
    #pragma unroll
    for (int v = 0; v < 8; ++v) {
        int r = row0 + half * 8 + v;             // C/D layout: VGPR v, lane-half selects M/M+8
        if (r < N) {
            float val = acc[v] + bv;
            long idx = (long)r * 64 + col0 + l16;
            outS[idx] = val;
            outAcc[idx] += val * w;
        }
    }
}

// ---------- link prediction: wave per pair, float2 per lane, shuffle reduce ----------
__global__ void gnn_link(const int* __restrict__ eli, const float* __restrict__ emb,
                         float* __restrict__ out, long EL) {
    long t = (long)blockIdx.x * blockDim.x + threadIdx.x;
    long e = t >> 5;
    int lane = (int)(t & 31);
    if (e >= EL) return;
    int a = eli[e], b = eli[EL + e];
    int f = lane * 2;
    float p = emb[(long)a * 64 + f]     * emb[(long)b * 64 + f]
            + emb[(long)a * 64 + f + 1] * emb[(long)b * 64 + f + 1];
    #pragma unroll
    for (int off = 16; off > 0; off >>= 1) p += __shfl_down(p, off, 32);
    if (lane == 0) out[e] = p;
}

// ---------- host launcher ----------
static inline int blks(long n, int b) { return (int)((n + b - 1) / b); }

extern "C" void kernel_launch(void* const* d_in, const int* in_sizes, int n_in,
                              void* d_out, int out_size, void* d_ws, size_t ws_size,
                              hipStream_t stream) {
    const float* x      = (const float*)d_in[0];
    const int*   ei     = (const int*)d_in[1];
    const int*   eli    = (const int*)d_in[2];
    const float* Wg     = (const float*)d_in[3];
    const float* att_s  = (const float*)d_in[4];
    const float* att_d  = (const float*)d_in[5];
    const float* b_gat  = (const float*)d_in[6];
    const float* W1l    = (const float*)d_in[7];
    const float* W1r    = (const float*)d_in[8];
    const float* b1     = (const float*)d_in[9];
    const float* W2l    = (const float*)d_in[10];
    const float* W2r    = (const float*)d_in[11];
    const float* b2     = (const float*)d_in[12];
    const float* alpha  = (const float*)d_in[13];

    const int  N  = in_sizes[0] / 11;
    const long E  = in_sizes[1] / 2;
    const long EL = in_sizes[2] / 2;
    const int* src = ei;
    const int* dst = ei + E;

    // workspace layout (floats)
    float* ws      = (float*)d_ws;
    float* wts     = ws;                 // 4
    float* wa_s    = ws + 4;             // 16
    float* wa_d    = ws + 20;            // 16
    float* a_src   = ws + 64;            // N
    float* a_dst   = a_src + N;          // N
    float* mkeyf   = a_dst + N;          // N (used as unsigned keys)
    float* ssum    = mkeyf + N;          // N
    float* deg     = ssum + N;           // N
    float* buf_h   = deg + N;            // N*64 : h = x@W, later agg/mean scratch
    float* buf_gat = buf_h + (long)N * 64;   // N*64 : GAT out, later sage2 out
    float* buf_s   = buf_gat + (long)N * 64; // N*64 : sage1 out
    float* buf_out = buf_s + (long)N * 64;   // N*64 : weighted accumulator

    const int B = 256;

    gnn_prep<<<1, 64, 0, stream>>>(alpha, Wg, att_s, att_d, wts, wa_s, wa_d);

    gnn_zero<<<1024, B, 0, stream>>>(mkeyf, (long)N);
    gnn_zero<<<1024, B, 0, stream>>>(ssum,  (long)N);
    gnn_zero<<<1024, B, 0, stream>>>(deg,   (long)N);
    gnn_zero<<<4096, B, 0, stream>>>(buf_gat, (long)N * 64);

    gnn_xw<<<blks((long)N * 64, B), B, 0, stream>>>(x, Wg, buf_h, N);
    gnn_attn<<<blks(N, B), B, 0, stream>>>(x, wa_s, wa_d, a_src, a_dst, N);

    gnn_seg_max<<<blks(E + N, B), B, 0, stream>>>(src, dst, a_src, a_dst,
                                                  (unsigned*)mkeyf, E, N);
    gnn_seg_sum<<<blks(E + N, B), B, 0, stream>>>(src, dst, a_src, a_dst,
                                                  (const unsigned*)mkeyf, ssum, E, N);
    gnn_gat_agg<<<blks((E + N) * 32, B), B, 0, stream>>>(src, dst, a_src, a_dst,
                                                         (const unsigned*)mkeyf, ssum,
                                                         buf_h, buf_gat, E, N);
    gnn_bias_seed<<<blks((long)N * 64, B), B, 0, stream>>>(buf_gat, b_gat, wts, buf_out, N);

    gnn_deg<<<blks(E, B), B, 0, stream>>>(dst, deg, E);

    // SAGE layer 1: input buf_gat
    gnn_zero<<<4096, B, 0, stream>>>(buf_h, (long)N * 64);
    gnn_sage_agg<<<blks(E * 32, B), B, 0, stream>>>(src, dst, buf_gat, buf_h, E);
    gnn_mean<<<blks((long)N * 64, B), B, 0, stream>>>(buf_h, deg, N);
    gnn_sage_gemm<<<blks(N, 16), 128, 0, stream>>>(buf_h, buf_gat, W1l, W1r, b1, wts,
                                                   2, buf_s, buf_out, N);

    // SAGE layer 2: input buf_s
    gnn_zero<<<4096, B, 0, stream>>>(buf_h, (long)N * 64);
    gnn_sage_agg<<<blks(E * 32, B), B, 0, stream>>>(src, dst, buf_s, buf_h, E);
    gnn_mean<<<blks((long)N * 64, B), B, 0, stream>>>(buf_h, deg, N);
    gnn_sage_gemm<<<blks(N, 16), 128, 0, stream>>>(buf_h, buf_s, W2l, W2r, b2, wts,
                                                   3, buf_gat, buf_out, N);

    // link prediction dot products
    gnn_link<<<blks(EL * 32, B), B, 0, stream>>>(eli, buf_out, (float*)d_out, EL);
}